// S4DKernel_60705067762283
// MI455X (gfx1250) — compile-verified
//
#include <hip/hip_runtime.h>
#include <math.h>

typedef float v2f __attribute__((ext_vector_type(2)));
typedef float v8f __attribute__((ext_vector_type(8)));

#define HDIM  1024
#define NHALF 32
#define LLEN  4096
#define AST   66   // padded LDS row stride (floats) to avoid bank conflicts

// One workgroup per h. 128 threads = 4 waves (wave32).
// Output tile is 64 (l1) x 64 (l2), l = 64*l1 + l2.
__global__ __launch_bounds__(128) void s4d_wmma_kernel(
    const float* __restrict__ C_ri,        // (H, 32, 2)
    const float* __restrict__ log_dt,      // (H,)
    const float* __restrict__ log_A_real,  // (H, 32)
    const float* __restrict__ A_imag,      // (H, 32)
    float* __restrict__ out)               // (H, 4096)
{
  __shared__ float sA[64 * AST];           // A[l1][k]
  __shared__ float sB[64 * AST];           // B[k][l2]
  __shared__ float s_ar[NHALF], s_ai[NHALF], s_lr[NHALF], s_li[NHALF];

  const int h   = blockIdx.x;
  const int tid = threadIdx.x;

  // ---- Phase 1: per-n scalars: dtA = dt*A, lhs = C*(exp(dtA)-1)/A ----
  if (tid < NHALF) {
    const int n  = tid;
    const float dt  = expf(log_dt[h]);
    const float Are = -expf(log_A_real[h * NHALF + n]);
    const float Aim = A_imag[h * NHALF + n];
    const float dre = dt * Are;
    const float dim = dt * Aim;
    float s, c;
    sincosf(dim, &s, &c);
    const float e   = expf(dre);
    const float wre = e * c, wim = e * s;         // exp(dtA)
    const float mag2 = Are * Are + Aim * Aim;
    const float nr = wre - 1.0f, ni = wim;        // exp(dtA)-1
    const float tre = (nr * Are + ni * Aim) / mag2;  // (exp(dtA)-1)/A
    const float tim = (ni * Are - nr * Aim) / mag2;
    const float Cre = C_ri[(h * NHALF + n) * 2 + 0];
    const float Cim = C_ri[(h * NHALF + n) * 2 + 1];
    s_ar[n] = dre;
    s_ai[n] = dim;
    s_lr[n] = Cre * tre - Cim * tim;
    s_li[n] = Cre * tim + Cim * tre;
  }
  __syncthreads();

  // ---- Phase 2: generate A (64x64) and B (64x64) in LDS ----
  // A[l1][k]: k<32 -> 2*Re(lhs * exp(dtA*64*l1)), k>=32 -> -2*Im(...)
  for (int idx = tid; idx < 64 * 64; idx += 128) {
    const int l1 = idx >> 6, k = idx & 63, n = k & 31;
    const float l  = (float)(l1 << 6);
    const float e  = expf(s_ar[n] * l);
    float s, c;
    sincosf(s_ai[n] * l, &s, &c);
    const float zre = e * c, zim = e * s;
    const float lr = s_lr[n], li = s_li[n];
    const float val = (k < 32) ? 2.0f * (lr * zre - li * zim)
                               : -2.0f * (lr * zim + li * zre);
    sA[l1 * AST + k] = val;
  }
  // B[k][l2]: k<32 -> Re(exp(dtA*l2)), k>=32 -> Im(exp(dtA*l2))
  for (int idx = tid; idx < 64 * 64; idx += 128) {
    const int k = idx >> 6, l2 = idx & 63, n = k & 31;
    const float l = (float)l2;
    const float e = expf(s_ar[n] * l);
    float s, c;
    sincosf(s_ai[n] * l, &s, &c);
    sB[k * AST + l2] = (k < 32) ? e * c : e * s;
  }
  __syncthreads();

  // ---- Phase 3: 64x64x64 GEMM via V_WMMA_F32_16X16X4_F32 ----
  const int wave   = tid >> 5;
  const int lane   = tid & 31;
  const int n16    = lane & 15;   // M index for A frag / N index for B,C frags
  const int khalf  = lane >> 4;   // which K-half of the 4-wide K block
  const int l1base = wave << 4;   // each wave owns 16 rows of the tile

  // Preload this wave's 16 A fragments (K blocks of 4).
  // ISA layout 16x4 f32 A: vgpr = k&1, lane-half = k>>1, lane%16 = M.
  v2f afrag[16];
#pragma unroll
  for (int kb = 0; kb < 16; ++kb) {
    const float* p = &sA[(l1base + n16) * AST + 4 * kb + 2 * khalf];
    v2f a; a.x = p[0]; a.y = p[1];
    afrag[kb] = a;
  }

  float* op = out + (size_t)h * LLEN;

  for (int j = 0; j < 4; ++j) {            // 4 column tiles of 16 (l2)
    v8f acc = {};
#pragma unroll
    for (int kb = 0; kb < 16; ++kb) {      // K = 64 in steps of 4
      const int k0 = 4 * kb + 2 * khalf;
      v2f b;
      b.x = sB[k0 * AST + 16 * j + n16];
      b.y = sB[(k0 + 1) * AST + 16 * j + n16];
      acc = __builtin_amdgcn_wmma_f32_16x16x4_f32(
          /*neg_a=*/false, afrag[kb], /*neg_b=*/false, b,
          /*c_mod=*/(short)0, acc, /*reuse_a=*/false, /*reuse_b=*/false);
    }
    // C/D layout: vgpr r, lane -> M = r + 8*(lane>>4), N = lane&15
#pragma unroll
    for (int r = 0; r < 8; ++r) {
      const int m = r + 8 * khalf;
      const int l = (l1base + m) * 64 + 16 * j + n16;
      op[l] = acc[r];
    }
  }
}

extern "C" void kernel_launch(void* const* d_in, const int* in_sizes, int n_in,
                              void* d_out, int out_size, void* d_ws, size_t ws_size,
                              hipStream_t stream) {
  const float* C_ri       = (const float*)d_in[0];
  const float* log_dt     = (const float*)d_in[1];
  const float* log_A_real = (const float*)d_in[2];
  const float* A_imag     = (const float*)d_in[3];
  // d_in[4] is L (scalar int) == 4096, baked in as LLEN.
  float* out = (float*)d_out;

  s4d_wmma_kernel<<<HDIM, 128, 0, stream>>>(C_ri, log_dt, log_A_real, A_imag, out);
}